// MipNerf_75720273428551
// MI455X (gfx1250) — compile-verified
//
#include <hip/hip_runtime.h>

typedef __attribute__((ext_vector_type(2))) float v2f;
typedef __attribute__((ext_vector_type(8))) float v8f;

#define TILE 16
#define THREADS 128

// P basis (3 x 21), row-major from the reference.
__constant__ float Pc[3][21] = {
    {0.8506508f, 0.0f, 0.5257311f, 0.809017f, 0.5f, 0.309017f, 0.5257311f,
     0.8506508f, 0.0f, 1.0f, 0.0f, 0.0f, 0.809017f, 0.5f, -0.309017f,
     0.8506508f, 0.0f, -0.5257311f, 0.309017f, 0.809017f, -0.5f},
    {0.0f, 0.5257311f, -0.8506508f, 0.5f, 0.309017f, -0.809017f, 0.0f, 1.0f,
     0.0f, -0.5257311f, 0.8506508f, 0.0f, -0.309017f, 0.809017f, -0.5f, 0.0f,
     0.5257311f, 0.8506508f, -0.309017f, 0.809017f, 0.5f},
    {0.309017f, 0.809017f, 0.5f, 0.5f, 0.309017f, 0.809017f, 0.5f, -0.309017f,
     0.809017f, 0.0f, 0.0f, 1.0f, -0.5f, 0.309017f, 0.809017f, -0.809017f,
     0.5f, 0.309017f, -0.809017f, 0.5f, -0.309017f}};

__device__ __forceinline__ void frustum_moments(float t0, float t1, float radius,
                                                float& t_mean, float& t_var,
                                                float& r_var) {
  float mu  = 0.5f * (t0 + t1);
  float hw  = 0.5f * (t1 - t0);
  float mu2 = mu * mu, hw2 = hw * hw;
  float hw4 = hw2 * hw2;
  float inv_denom = 1.0f / (3.0f * mu2 + hw2);
  t_mean = mu + 2.0f * mu * hw2 * inv_denom;
  t_var  = hw2 * (1.0f / 3.0f) -
           (4.0f / 15.0f) * (hw4 * (12.0f * mu2 - hw2) * inv_denom * inv_denom);
  r_var  = radius * radius *
           (0.25f * mu2 + (5.0f / 12.0f) * hw2 - (4.0f / 15.0f) * hw4 * inv_denom);
}

__device__ __forceinline__ v8f wmma_f32_k4(v2f a, v2f b, v8f c) {
  // V_WMMA_F32_16X16X4_F32: D = A(16x4) * B(4x16) + C(16x16)
  return __builtin_amdgcn_wmma_f32_16x16x4_f32(
      /*neg_a=*/false, a, /*neg_b=*/false, b,
      /*c_mod=*/(short)0, c, /*reuse_a=*/false, /*reuse_b=*/false);
}

// 2^deg as float via exponent field (deg in [0,30])
__device__ __forceinline__ float pow2i(int deg) {
  union { unsigned u; float f; } v;
  v.u = (unsigned)(127 + deg) << 23;
  return v.f;
}

__global__ __launch_bounds__(THREADS) void mipnerf_ipe_kernel(
    const float* __restrict__ ray_o, const float* __restrict__ ray_d,
    const float* __restrict__ fg_z, const float* __restrict__ bg_z,
    const float* __restrict__ radii, float* __restrict__ out,
    int Nrays, int Ns) {
  // LDS staging for one 16-sample tile.
  __shared__ float sA[TILE][4];    // contracted mean (K padded 3->4), WMMA A
  __shared__ float sC[TILE][12];   // contracted cov flat (K padded 9->12), WMMA A
  __shared__ float sYb[TILE][22];  // y  = mean_c . P   (21 cols, padded)
  __shared__ float sYv[TILE][22];  // yv = p_k^T C p_k
  __shared__ float sFgM[TILE][3];  // fg mean
  __shared__ float sFgV[TILE][3];  // fg diagonal cov
  // Zero-padded WMMA B tables (built once per block; indexed [col32][k]).
  __shared__ float sBY[32][4];     // P:   sBY[c][k] = (c<21 && k<3) ? P[k][c] : 0
  __shared__ float sBV[32][12];    // Q:   sBV[c][j] = (c<21 && j<9) ? P[j/3][c]*P[j%3][c] : 0

  const int  tid          = threadIdx.x;
  const long tileBase     = (long)blockIdx.x * TILE;
  const long totalSamples = (long)Nrays * Ns;
  const int  Nz           = Ns + 1;

  // ---- Phase 0: build padded B tables in LDS (cheap, one-time per block) ----
  {
    // sBY: 128 entries, one per thread
    int c = tid >> 2, k = tid & 3;
    sBY[c][k] = (c < 21 && k < 3) ? Pc[k][c] : 0.0f;
    // sBV: 384 entries, 3 per thread
    for (int idx = tid; idx < 32 * 12; idx += THREADS) {
      int cc = idx / 12, j = idx - cc * 12;
      sBV[cc][j] = (cc < 21 && j < 9) ? Pc[j / 3][cc] * Pc[j % 3][cc] : 0.0f;
    }
  }

  // ---- Phase 1: per-sample scalar setup (threads 0..15) ----
  if (tid < TILE) {
    long sid = tileBase + tid;
    if (sid < totalSamples) {
      int n = (int)(sid / Ns);
      int s = (int)(sid - (long)n * Ns);

      float o3[3] = {ray_o[3 * n + 0], ray_o[3 * n + 1], ray_o[3 * n + 2]};
      float d3[3] = {ray_d[3 * n + 0], ray_d[3 * n + 1], ray_d[3 * n + 2]};
      float rad   = radii[n];
      float dmag  = fmaxf(1e-8f, d3[0] * d3[0] + d3[1] * d3[1] + d3[2] * d3[2]);
      float inv_dmag = 1.0f / dmag;

      // fg: diagonal covariance path
      {
        float t0 = fg_z[(long)n * Nz + s];
        float t1 = fg_z[(long)n * Nz + s + 1];
        float tm, tv, rv;
        frustum_moments(t0, t1, rad, tm, tv, rv);
#pragma unroll
        for (int c = 0; c < 3; ++c) {
          float dod    = d3[c] * d3[c];
          sFgM[tid][c] = d3[c] * tm + o3[c];
          sFgV[tid][c] = tv * dod + rv * (1.0f - dod * inv_dmag);
        }
      }

      // bg: full covariance + scene contraction
      {
        float t0 = bg_z[(long)n * Nz + s];
        float t1 = bg_z[(long)n * Nz + s + 1];
        float tm, tv, rv;
        frustum_moments(t0, t1, rad, tm, tv, rv);

        float mb[3];
#pragma unroll
        for (int c = 0; c < 3; ++c) mb[c] = d3[c] * tm + o3[c];

        float C[3][3];
#pragma unroll
        for (int a = 0; a < 3; ++a)
#pragma unroll
          for (int b = 0; b < 3; ++b) {
            float dd = d3[a] * d3[b];
            C[a][b]  = tv * dd + rv * (((a == b) ? 1.0f : 0.0f) - dd * inv_dmag);
          }

        float r2   = mb[0] * mb[0] + mb[1] * mb[1] + mb[2] * mb[2];
        float rn   = sqrtf(r2);
        float nn   = rn + 1e-6f;
        float invn = 1.0f / nn;
        float aJ   = (2.0f - invn) * invn;  // 2/n - 1/n^2 (contraction factor too)
        float bJ   = invn * invn - aJ;      // a + b = 1/n^2

        sA[tid][0] = aJ * mb[0];
        sA[tid][1] = aJ * mb[1];
        sA[tid][2] = aJ * mb[2];
        sA[tid][3] = 0.0f;

        float invr = 1.0f / fmaxf(rn, 1e-12f);
        float u[3] = {mb[0] * invr, mb[1] * invr, mb[2] * invr};
        float J[3][3];
#pragma unroll
        for (int a = 0; a < 3; ++a)
#pragma unroll
          for (int b = 0; b < 3; ++b)
            J[a][b] = bJ * u[a] * u[b] + ((a == b) ? aJ : 0.0f);

        // Cc = J * C * J  (J symmetric)
        float M[3][3];
#pragma unroll
        for (int a = 0; a < 3; ++a)
#pragma unroll
          for (int b = 0; b < 3; ++b)
            M[a][b] = J[a][0] * C[0][b] + J[a][1] * C[1][b] + J[a][2] * C[2][b];
#pragma unroll
        for (int a = 0; a < 3; ++a)
#pragma unroll
          for (int b = 0; b < 3; ++b)
            sC[tid][a * 3 + b] =
                M[a][0] * J[0][b] + M[a][1] * J[1][b] + M[a][2] * J[2][b];
        sC[tid][9] = sC[tid][10] = sC[tid][11] = 0.0f;
      }
    } else {
      // Keep WMMA inputs defined for a (never-taken with these sizes) ragged tile.
#pragma unroll
      for (int k = 0; k < 4; ++k) sA[tid][k] = 0.0f;
#pragma unroll
      for (int k = 0; k < 12; ++k) sC[tid][k] = 0.0f;
#pragma unroll
      for (int c = 0; c < 3; ++c) { sFgM[tid][c] = 0.0f; sFgV[tid][c] = 0.0f; }
    }
  }
  __syncthreads();

  // ---- Phase 2: WMMA projections through P ----
  // f32 A 16x4 layout: lane l holds A[l&15][2*(l>>4) + r] in VGPR r.
  // B 4x16 mirrored:  lane l holds B[2*(l>>4) + r][l&15].
  // D 16x16:          lane l, VGPR r holds D[r + 8*(l>>4)][l&15].
  const int wave = tid >> 5;
  const int lane = tid & 31;
  const int m    = lane & 15;
  const int half = lane >> 4;
  const int koff = 2 * half;

  if (wave == 0) {
    // yb = mean_c (16x4, K=3 used) * P (two 4x16 tiles) -- fragments via ds_load_b64
    v2f a  = *(const v2f*)&sA[m][koff];
    v2f b0 = *(const v2f*)&sBY[m][koff];       // tile 0: cols 0..15
    v2f b1 = *(const v2f*)&sBY[16 + m][koff];  // tile 1: cols 16..20 (+pad)
    v8f c0 = {};
    v8f c1 = {};
    c0 = wmma_f32_k4(a, b0, c0);
    c1 = wmma_f32_k4(a, b1, c1);
#pragma unroll
    for (int r = 0; r < 8; ++r) sYb[r + 8 * half][m] = c0[r];
    if (m < 5) {
#pragma unroll
      for (int r = 0; r < 8; ++r) sYb[r + 8 * half][16 + m] = c1[r];
    }
  } else if (wave == 1 || wave == 2) {
    // yv = Cflat(16x9) * Q(9x16), chained K=4 WMMAs; all fragments preloaded.
    int t     = wave - 1;
    int col32 = t * 16 + m;
    v2f a0 = *(const v2f*)&sC[m][0 + koff];
    v2f a1 = *(const v2f*)&sC[m][4 + koff];
    v2f a2 = *(const v2f*)&sC[m][8 + koff];
    v2f b0 = *(const v2f*)&sBV[col32][0 + koff];
    v2f b1 = *(const v2f*)&sBV[col32][4 + koff];
    v2f b2 = *(const v2f*)&sBV[col32][8 + koff];
    v8f c = {};
    c = wmma_f32_k4(a0, b0, c);
    c = wmma_f32_k4(a1, b1, c);
    c = wmma_f32_k4(a2, b2, c);
    if (col32 < 21) {
#pragma unroll
      for (int r = 0; r < 8; ++r) sYv[r + 8 * half][col32] = c[r];
    }
  }
  __syncthreads();

  // ---- Phase 3: IPE transcendentals + streamed NT stores ----
  // Per sample: 384 (y, var) pairs; sin half + cos half share one exp.
  // Layout per sample (768 floats): [fg_sin 48 | fg_cos 48 | bg_sin 336 | bg_cos 336]
  for (int i = 0; i < TILE; ++i) {
    long sid = tileBase + i;
    if (sid >= totalSamples) break;
    size_t base = (size_t)sid * 768;
#pragma unroll
    for (int j = 0; j < 3; ++j) {
      int q = tid + j * THREADS;  // 0..383; j>0 => q>=128 => bg path only
      float y, var;
      int   fs, fc;
      if (q < 48) {
        int   deg = q / 3;
        int   c   = q - deg * 3;
        float sc  = pow2i(deg);
        y   = sFgM[i][c] * sc;
        var = sFgV[i][c] * pow2i(2 * deg);
        fs  = q;
        fc  = q + 48;
      } else {
        int   g   = q - 48;
        int   deg = g / 21;
        int   k   = g - deg * 21;
        float sc  = pow2i(deg);
        y   = sYb[i][k] * sc;
        var = sYv[i][k] * pow2i(2 * deg);
        fs  = 96 + g;
        fc  = 96 + 336 + g;
      }
      float e = __expf(-0.5f * var);
      __builtin_nontemporal_store(e * __sinf(y), &out[base + fs]);
      __builtin_nontemporal_store(e * __cosf(y), &out[base + fc]);
    }
  }
}

extern "C" void kernel_launch(void* const* d_in, const int* in_sizes, int n_in,
                              void* d_out, int out_size, void* d_ws,
                              size_t ws_size, hipStream_t stream) {
  const float* ray_o = (const float*)d_in[0];
  const float* ray_d = (const float*)d_in[1];
  const float* fg_z  = (const float*)d_in[2];
  const float* bg_z  = (const float*)d_in[3];
  const float* radii = (const float*)d_in[4];
  float*       out   = (float*)d_out;

  int  Nrays  = in_sizes[4];              // radii has N elements
  int  Ns     = in_sizes[2] / Nrays - 1;  // z_vals have N*(Ns+1)
  long total  = (long)Nrays * Ns;         // 131072 samples
  int  blocks = (int)((total + TILE - 1) / TILE);

  mipnerf_ipe_kernel<<<blocks, THREADS, 0, stream>>>(ray_o, ray_d, fg_z, bg_z,
                                                     radii, out, Nrays, Ns);
}